// timeGap2_55018531062157
// MI455X (gfx1250) — compile-verified
//
#include <hip/hip_runtime.h>
#include <stdint.h>

// ---------------------------------------------------------------------------
// 4 embedding tables [100, 256] f32; indices [128,500] i32 per table; output =
// 4 concatenated [64000, 256] row gathers = 262 MB of write-once stores.
// Roofline: 262 MB / 23.3 TB/s ~= 11.3 us -> pure store-BW bound.
// Plan: TDM-stage each 100KB table into LDS once per block (L2-resident reads,
// ~20% of output volume), then scatter gathered rows LDS->HBM with async B128
// stores (no VGPR round trip, ASYNCcnt-deep pipeline, NT so the 262MB stream
// doesn't evict the 192MB L2).
// ---------------------------------------------------------------------------
#define EMB           256
#define NROWS         100
#define BS            (128 * 500)          // 64000 rows per table
#define ROW_BYTES     (EMB * 4)            // 1024 B per gathered row
#define TABLE_ELEMS   (NROWS * EMB)        // 25600 f32
#define TABLE_BYTES   (TABLE_ELEMS * 4)    // 102400 B (fits 320KB WGP LDS)
#define THREADS       256                  // 8 wave32s
#define WAVES         8
#define ROWS_PER_BLOCK 512
#define ROWS_PER_WAVE  (ROWS_PER_BLOCK / WAVES)     // 64 contiguous rows/wave
#define BLOCKS_PER_TABLE (BS / ROWS_PER_BLOCK)      // 125 (exact)

typedef uint32_t v4u __attribute__((ext_vector_type(4)));
typedef int      v8i __attribute__((ext_vector_type(8)));
typedef int      v4i __attribute__((ext_vector_type(4)));

// ---- CDNA5 async LDS->global store (VGLOBAL encoding, ASYNCcnt-tracked) ----
// ISA 15.18.3: global_store_async_from_lds_b128 vaddr, vsrc(lds), off.
// INST_OFFSET is added to BOTH global and LDS addresses (ISA §10/§15.18), so
// offset:512 advances both sides for the second half of a 1KB row.
// th:TH_STORE_NT: output is streaming write-once > L2 capacity.
__device__ __forceinline__ void async_store_b128_nt(void* gaddr, uint32_t lds_off) {
    asm volatile("global_store_async_from_lds_b128 %0, %1, off th:TH_STORE_NT"
                 :: "v"(gaddr), "v"(lds_off) : "memory");
}
__device__ __forceinline__ void async_store_b128_nt_o512(void* gaddr, uint32_t lds_off) {
    asm volatile("global_store_async_from_lds_b128 %0, %1, off offset:512 th:TH_STORE_NT"
                 :: "v"(gaddr), "v"(lds_off) : "memory");
}
__device__ __forceinline__ void wait_asynccnt0() {
    asm volatile("s_wait_asynccnt 0" ::: "memory");
}

// Generic (flat) shared pointer -> 32-bit wave-relative LDS offset.
// addrspacecast(3->flat) = {aperture_hi, lds_offset_lo}; truncate.
__device__ __forceinline__ uint32_t lds_offset_of(const void* p) {
    return (uint32_t)(uintptr_t)p;
}

// ---- TDM: stage one whole embedding table (100KB) into LDS as one DMA op ---
// D# per CDNA5 ISA §8.3/8.4: count=1, type=2, data_size=4B, tensor = one line
// of 25600 elems, tile = whole line. No multicast/padding/iterate/gather.
__device__ __forceinline__ void tdm_load_table_to_lds(const float* W, uint32_t lds_addr) {
    const uint64_t ga = (uint64_t)(uintptr_t)W;
    const uint64_t D0 = (uint64_t)TABLE_ELEMS;   // 25600, fits tile_dim0[15:0]

    v4u g0;
    g0.x = 1u;                                   // [1:0] count=1 (valid), user mode
    g0.y = lds_addr;                             // [63:32] lds_addr (bytes)
    g0.z = (uint32_t)ga;                         // [95:64] global_addr[31:0]
    g0.w = (uint32_t)((ga >> 32) & 0x01FFFFFFu)  // [120:96] global_addr[56:32]
         | (2u << 30);                           // [127:126] type=2

    // group1 as four LE 64-bit quads = bits [63:0],[127:64],[191:128],[255:192]
    uint64_t q0 = (2ull << 16)                       // data_size=2 -> 4 bytes
                | ((D0 & 0xFFFFull) << 48);          // tensor_dim0[15:0] @48
    uint64_t q1 = (D0 >> 16)                         // tensor_dim0[31:16]
                | (1ull << 16)                       // tensor_dim1=1 (bits 111:80)
                | (D0 << 48);                        // tile_dim0=25600 (bits 127:112)
    uint64_t q2 = 1ull                               // tile_dim1=1 (bits 143:128)
                | (D0 << 32);                        // tensor_dim0_stride[31:0] (bits 207:160)
    uint64_t q3 = (D0 << 16);                        // tensor_dim1_stride (bits 255:208)

    v8i g1 = { (int)(uint32_t)q0, (int)(uint32_t)(q0 >> 32),
               (int)(uint32_t)q1, (int)(uint32_t)(q1 >> 32),
               (int)(uint32_t)q2, (int)(uint32_t)(q2 >> 32),
               (int)(uint32_t)q3, (int)(uint32_t)(q3 >> 32) };
    v4i g2 = { 0, 0, 0, 0 };                         // dims 2/3 unused
    v4i g3 = { 0, 0, 0, 0 };
    v8i g4 = { 0, 0, 0, 0, 0, 0, 0, 0 };             // 6-arg toolchain extra group

    __builtin_amdgcn_tensor_load_to_lds(g0, g1, g2, g3, g4, /*cpol=*/0);
    __builtin_amdgcn_s_wait_tensorcnt(0);
}

// ---------------------------------------------------------------------------
// One block <-> one (table, 512-row chunk); each wave owns 64 contiguous rows
// (a contiguous 64KB store region). Phase 1: TDM-stage table. Phase 2: two
// coalesced lane loads fetch all 64 indices; v_readlane broadcasts each row's
// index; two async NT B128 stores per row stream the row from LDS to HBM.
// ---------------------------------------------------------------------------
__global__ void __launch_bounds__(THREADS)
timegap_gather_kernel(const int* __restrict__ i0, const int* __restrict__ i1,
                      const int* __restrict__ i2, const int* __restrict__ i3,
                      const float* __restrict__ w0, const float* __restrict__ w1,
                      const float* __restrict__ w2, const float* __restrict__ w3,
                      float* __restrict__ out) {
    extern __shared__ float lds_tab[];   // TABLE_BYTES

    const int t     = blockIdx.x / BLOCKS_PER_TABLE;
    const int chunk = blockIdx.x % BLOCKS_PER_TABLE;

    const int*   idx = (t == 0) ? i0 : (t == 1) ? i1 : (t == 2) ? i2 : i3;
    const float* W   = (t == 0) ? w0 : (t == 1) ? w1 : (t == 2) ? w2 : w3;
    float* outT = out + (size_t)t * (size_t)BS * EMB;

    const uint32_t lds_base = lds_offset_of(&lds_tab[0]);

    // Phase 1: first wave issues the table DMA (TDM ignores EXEC; descriptor
    // is wave-uniform -> SGPRs), waits TENSORcnt; barrier publishes to all.
    if (threadIdx.x < 32) {
        tdm_load_table_to_lds(W, lds_base);
    }
    __syncthreads();

    // Phase 2: gather-scatter 64 contiguous rows per wave.
    const int lane = threadIdx.x & 31;
    const int wave = threadIdx.x >> 5;
    const int base = chunk * ROWS_PER_BLOCK + wave * ROWS_PER_WAVE;

    // Preload this wave's 64 indices with two coalesced B32 loads.
    const int id_lo = idx[base + lane];
    const int id_hi = idx[base + 32 + lane];

    const uint32_t lane16 = (uint32_t)lane * 16u;
    float* dst0 = outT + (size_t)base * EMB + lane * 4;   // lane*16 bytes

    #pragma unroll 8
    for (int k = 0; k < ROWS_PER_WAVE; ++k) {
        // k is wave-uniform -> legal dynamic lane select for v_readlane.
        const int id = (k < 32) ? __builtin_amdgcn_readlane(id_lo, k)
                                : __builtin_amdgcn_readlane(id_hi, k - 32);
        const uint32_t src = lds_base + (uint32_t)id * ROW_BYTES + lane16;
        float* dst = dst0 + (size_t)k * EMB;
        async_store_b128_nt(dst, src);
        async_store_b128_nt_o512(dst, src);
    }

    // Drain outstanding async stores before wave exit (S_ENDPGM also implies
    // wait-idle; keep the dependency explicit).
    wait_asynccnt0();
}

extern "C" void kernel_launch(void* const* d_in, const int* in_sizes, int n_in,
                              void* d_out, int out_size, void* d_ws, size_t ws_size,
                              hipStream_t stream) {
    const int*   rgap    = (const int*)d_in[0];
    const int*   sgap    = (const int*)d_in[1];
    const int*   pcount  = (const int*)d_in[2];
    const int*   prcount = (const int*)d_in[3];
    const float* Wr      = (const float*)d_in[4];
    const float* Ws      = (const float*)d_in[5];
    const float* Wp      = (const float*)d_in[6];
    const float* Wpr     = (const float*)d_in[7];
    float* out = (float*)d_out;

    dim3 grid(4 * BLOCKS_PER_TABLE);   // 500 blocks, 4000 wave32s
    timegap_gather_kernel<<<grid, THREADS, TABLE_BYTES, stream>>>(
        rgap, sgap, pcount, prcount, Wr, Ws, Wp, Wpr, out);
}